// BaseSmear_87127706567258
// MI455X (gfx1250) — compile-verified
//
#include <hip/hip_runtime.h>

typedef float v2f __attribute__((ext_vector_type(2)));
typedef float v8f __attribute__((ext_vector_type(8)));

#define NCAM 4
#define NCH  32
#define IMH  480
#define IMW  640
#define CPC  (NCH + 5)   // 37 channels per camera in output grid

__device__ __forceinline__ int iclamp(int x, int lo, int hi) {
    return x < lo ? lo : (x > hi ? hi : x);
}

// One wave handles 16 voxels. WMMA 16x16x4 f32 computes the 3x4 projection for
// all 4 cameras of those 16 voxels in one instruction:
//   A (16x4): rows 0-5  = cam0,cam1 projection rows; rows 8-13 = cam2,cam3; rest 0
//   B (4x16): column n  = [x, y, z, 1] of voxel n
//   D (16x16): lane n (n<16) holds rows 0-7 -> cams 0,1 of voxel n in D[0..5]
//              lane n+16    holds rows 8-15 -> cams 2,3 of voxel n in D[0..5]
__global__ __launch_bounds__(256) void smear_wmma_kernel(
    const int*   __restrict__ gsz,     // [3]
    const float* __restrict__ T0w,     // [16] row-major 4x4
    const float* __restrict__ center,  // [3]
    const float* __restrict__ pitchp,  // [1]
    const float* __restrict__ images,  // [I,C,H,W]
    const float* __restrict__ P,       // [I,3,4]
    const float* __restrict__ Tcw,     // [I,4,4]
    float*       __restrict__ out,     // [I*CPC*N + 3*N]
    int N)
{
    const int Ygrid = gsz[1], Zgrid = gsz[2];
    const int lane = threadIdx.x & 31;
    const int half = lane >> 4;                       // 0: cams 0-1, 1: cams 2-3
    const int wave = (int)((blockIdx.x * blockDim.x + threadIdx.x) >> 5);
    int vn = wave * 16 + (lane & 15);
    if (vn >= N) vn = N - 1;       // keep EXEC all-ones for WMMA (dup stores OK)

    // ---- voxel -> world point ------------------------------------------------
    int k  = vn % Zgrid;
    int tq = vn / Zgrid;
    int j  = tq % Ygrid;
    int i  = tq / Ygrid;

    const float pitch = pitchp[0];
    float idxf[3] = { (float)i, (float)j, (float)k };
    float p0[3];
    #pragma unroll
    for (int d = 0; d < 3; ++d)
        p0[d] = center[d] + pitch * (idxf[d] - ((float)gsz[d] - 1.0f) * 0.5f);

    float q0 = p0[0] - T0w[3], q1 = p0[1] - T0w[7], q2 = p0[2] - T0w[11];
    // pts_w = R0^T * (p0 - t0)
    float wx = T0w[0]*q0 + T0w[4]*q1 + T0w[8]*q2;
    float wy = T0w[1]*q0 + T0w[5]*q1 + T0w[9]*q2;
    float wz = T0w[2]*q0 + T0w[6]*q1 + T0w[10]*q2;

    // ---- build A (camera rows) and B (homogeneous points), run WMMA ----------
    int row = lane & 15;
    float a0 = 0.0f, a1 = 0.0f;
    {
        int cam = -1, prow = 0;
        if (row < 6)                  { cam = row / 3;          prow = row % 3;       }
        else if (row >= 8 && row < 14){ cam = 2 + (row - 8) / 3; prow = (row - 8) % 3; }
        if (cam >= 0) {
            const float* pr = P + cam * 12 + prow * 4;
            a0 = pr[half ? 2 : 0];    // VGPR0 holds K0 (lanes 0-15) / K2 (lanes 16-31)
            a1 = pr[half ? 3 : 1];    // VGPR1 holds K1 / K3
        }
    }
    v2f A; A[0] = a0; A[1] = a1;
    v2f B; B[0] = half ? wz : wx;     // B VGPR0 = row K0 / K2
           B[1] = half ? 1.0f : wy;   // B VGPR1 = row K1 / K3
    v8f Cz = {};
    v8f D = __builtin_amdgcn_wmma_f32_16x16x4_f32(
        false, A, false, B, (short)0, Cz, false, false);

    // ---- per-lane: 2 cameras of this voxel -----------------------------------
    const size_t Ns = (size_t)N;
    #pragma unroll
    for (int t = 0; t < 2; ++t) {
        const int ci = half * 2 + t;
        float px = D[t*3 + 0], py = D[t*3 + 1], pz = D[t*3 + 2];
        float u = px / pz;
        float v = py / pz;
        float valid = (u >= 0.0f && u <= (float)(IMW - 1) &&
                       v >= 0.0f && v <= (float)(IMH - 1) && pz > 0.0f) ? 1.0f : 0.0f;

        float x0f = floorf(u), y0f = floorf(v);
        float fx = u - x0f,    fy = v - y0f;
        int x0 = (int)x0f, y0 = (int)y0f;
        int x1 = x0 + 1,   y1 = y0 + 1;
        float m00 = (x0 >= 0 && x0 < IMW && y0 >= 0 && y0 < IMH) ? 1.0f : 0.0f;
        float m10 = (x1 >= 0 && x1 < IMW && y0 >= 0 && y0 < IMH) ? 1.0f : 0.0f;
        float m01 = (x0 >= 0 && x0 < IMW && y1 >= 0 && y1 < IMH) ? 1.0f : 0.0f;
        float m11 = (x1 >= 0 && x1 < IMW && y1 >= 0 && y1 < IMH) ? 1.0f : 0.0f;
        int x0c = iclamp(x0, 0, IMW - 1), x1c = iclamp(x1, 0, IMW - 1);
        int y0c = iclamp(y0, 0, IMH - 1), y1c = iclamp(y1, 0, IMH - 1);
        float w00 = (1.0f - fx) * (1.0f - fy) * m00;
        float w10 = fx * (1.0f - fy) * m10;
        float w01 = (1.0f - fx) * fy * m01;
        float w11 = fx * fy * m11;

        const float* imb = images + (size_t)ci * NCH * IMH * IMW;
        const size_t o00 = (size_t)y0c * IMW + x0c;
        const size_t o10 = (size_t)y0c * IMW + x1c;
        const size_t o01 = (size_t)y1c * IMW + x0c;
        const size_t o11 = (size_t)y1c * IMW + x1c;

        float* ob = out + (size_t)ci * CPC * Ns + (size_t)vn;
        // bilinear taps hit L2 (192MB holds all 157MB of images); output stores
        // are non-temporal so the streaming 158MB write doesn't evict them.
        #pragma unroll 4
        for (int c = 0; c < NCH; ++c) {
            const float* ip = imb + (size_t)c * (IMH * IMW);
            float s = w00 * ip[o00] + w10 * ip[o10] + w01 * ip[o01] + w11 * ip[o11];
            __builtin_nontemporal_store(s, ob + (size_t)c * Ns);
        }
        __builtin_nontemporal_store(pz,    ob + (size_t)NCH * Ns);
        __builtin_nontemporal_store(valid, ob + (size_t)(NCH + 1) * Ns);

        // dirs = normalize(pts_w - cam_center);  cam_c = -R^T t
        const float* T = Tcw + ci * 16;
        float c0 = -(T[0]*T[3] + T[4]*T[7] + T[8]*T[11]);
        float c1 = -(T[1]*T[3] + T[5]*T[7] + T[9]*T[11]);
        float c2 = -(T[2]*T[3] + T[6]*T[7] + T[10]*T[11]);
        float dx = wx - c0, dy = wy - c1, dz = wz - c2;
        float nrm = sqrtf(dx*dx + dy*dy + dz*dz);
        __builtin_nontemporal_store(dx / nrm, ob + (size_t)(NCH + 2) * Ns);
        __builtin_nontemporal_store(dy / nrm, ob + (size_t)(NCH + 3) * Ns);
        __builtin_nontemporal_store(dz / nrm, ob + (size_t)(NCH + 4) * Ns);
    }

    // coordinates output (one copy per voxel: low half-wave only)
    if (half == 0) {
        float* cb = out + (size_t)NCAM * CPC * Ns;
        __builtin_nontemporal_store(wx, cb + 0 * Ns + vn);
        __builtin_nontemporal_store(wy, cb + 1 * Ns + vn);
        __builtin_nontemporal_store(wz, cb + 2 * Ns + vn);
    }
}

extern "C" void kernel_launch(void* const* d_in, const int* in_sizes, int n_in,
                              void* d_out, int out_size, void* d_ws, size_t ws_size,
                              hipStream_t stream) {
    (void)in_sizes; (void)n_in; (void)d_ws; (void)ws_size;
    // out = I*(C+5)*N + 3*N = 151*N with I=4, C=32 (fixed by setup)
    const int N = out_size / (NCAM * CPC + 3);
    const int waves = (N + 15) / 16;          // 16 voxels per wave
    const long threads = (long)waves * 32;
    const int block = 256;
    const int grid = (int)((threads + block - 1) / block);
    smear_wmma_kernel<<<grid, block, 0, stream>>>(
        (const int*)  d_in[0],   // grid_size
        (const float*)d_in[1],   // T_0w
        (const float*)d_in[2],   // center
        (const float*)d_in[3],   // pitch
        (const float*)d_in[4],   // images
        (const float*)d_in[5],   // transformations (P)
        (const float*)d_in[6],   // T_cw
        (float*)d_out, N);
}